// DDConv_3D_12421045420308
// MI455X (gfx1250) — compile-verified
//
#include <hip/hip_runtime.h>
#include <math.h>

// ---------------------------------------------------------------------------
// DDConv3D forward for gfx1250 (MI455X). f32 WMMA (16x16x4) implicit GEMMs,
// LDS-staged B tiles, dual accumulator chains.
// ---------------------------------------------------------------------------

typedef float v2f __attribute__((ext_vector_type(2)));
typedef float v8f __attribute__((ext_vector_type(8)));

#define BATCH 2
#define CIN   32
#define OUTC  64
#define SDIM  28
#define SP    (28*28*28)   // 21952 spatial positions
#define XPVOL (30*30*30)   // 27000 padded volume
#define KDIM  (32*27)      // 864 GEMM K
#define N27   27
#define E8    8

// ---------------- pooled mean of x: [B,C] ----------------------------------
__global__ void pool_x_kernel(const float* __restrict__ x, float* __restrict__ pooled) {
    int t = threadIdx.x;               // 0..63
    const float* p = x + (size_t)t * SP;
    float s = 0.f;
    for (int i = 0; i < SP; i += 4) {
        float4 v = *(const float4*)(p + i);
        s += v.x + v.y + v.z + v.w;
    }
    pooled[t] = s * (1.f / (float)SP);
}

// ---------------- routing: r = sigmoid(pooled @ fw.T + fb) -----------------
__global__ void route_kernel(const float* __restrict__ pooled,
                             const float* __restrict__ fw,
                             const float* __restrict__ fb,
                             float* __restrict__ r) {
    int t = threadIdx.x;               // 0..15
    int b = t >> 3, e = t & 7;
    float z = fb[e];
    for (int c = 0; c < CIN; ++c) z += pooled[b * CIN + c] * fw[e * CIN + c];
    r[t] = 1.f / (1.f + expf(-z));
}

// ---------------- expert mix: kern[b] = sum_e r[b,e] * W[e] ----------------
__global__ void mix_kernel(const float* __restrict__ W, const float* __restrict__ r,
                           float* __restrict__ kern, int M) {
    int idx = blockIdx.x * blockDim.x + threadIdx.x;
    int per_b = M * KDIM;
    if (idx >= BATCH * per_b) return;
    int b = idx / per_b;
    int mk = idx - b * per_b;
    float s = 0.f;
#pragma unroll
    for (int e = 0; e < E8; ++e)
        s += r[b * E8 + e] * W[(size_t)e * per_b + mk];
    kern[idx] = s;
}

// ---------------- zero-pad x by 1 on each spatial side ---------------------
__global__ void pad_kernel(const float* __restrict__ x, float* __restrict__ xp) {
    int idx = blockIdx.x * blockDim.x + threadIdx.x;
    if (idx >= BATCH * CIN * XPVOL) return;
    int bc  = idx / XPVOL;
    int rem = idx - bc * XPVOL;
    int h = rem / 900, w = (rem / 30) % 30, d = rem % 30;
    float v = 0.f;
    if (h >= 1 && h <= SDIM && w >= 1 && w <= SDIM && d >= 1 && d <= SDIM)
        v = x[(size_t)bc * SP + (h - 1) * 784 + (w - 1) * 28 + (d - 1)];
    xp[idx] = v;
}

// ---------------- offset conv: LDS-staged implicit im2col GEMM -------------
// D[81 x 21952] = kern_p[81 x 864] * im2col(xp)[864 x 21952] + bp
// 6 waves per block (one per 16-row M tile); B tile staged once in LDS.
__global__ void __launch_bounds__(192)
offset_gemm_kernel(const float* __restrict__ kern_p, const float* __restrict__ xp,
                   const float* __restrict__ bp, float* __restrict__ offs) {
    __shared__ float Bt[KDIM * 16];          // 55.3 KB: B[k][col_local]
    int tid  = threadIdx.x;
    int lane = tid & 31, mt = tid >> 5;      // wave -> M tile 0..5 (M=81 pad 96)
    int ln = lane & 15, lh = lane >> 4;
    int colbase = blockIdx.x * 16;
    int b = blockIdx.z;
    const float* Xb = xp + (size_t)b * CIN * XPVOL;

    // Cooperative fill of B tile (implicit im2col, pad=1 folded into xp coords)
    for (int e = tid; e < KDIM * 16; e += 192) {
        int k = e >> 4, cl = e & 15;
        int col = colbase + cl;
        int oh = col / 784, ow = (col / 28) % 28, od = col % 28;
        int c = k / 27, rem = k - c * 27;
        int ka = rem / 9, kbt = (rem / 3) % 3, kc = rem % 3;
        Bt[e] = Xb[c * XPVOL + (oh + ka) * 900 + (ow + kbt) * 30 + (od + kc)];
    }
    __syncthreads();

    int mrow = mt * 16 + ln;
    int mclamp = mrow > 80 ? 80 : mrow;      // clamp A row; bogus rows never stored
    const float* Arow = kern_p + (size_t)(b * 81 + mclamp) * KDIM;

    v8f acc0 = {0.f, 0.f, 0.f, 0.f, 0.f, 0.f, 0.f, 0.f};
    v8f acc1 = {0.f, 0.f, 0.f, 0.f, 0.f, 0.f, 0.f, 0.f};
#pragma unroll 4
    for (int kb = 0; kb < KDIM; kb += 8) {
        int k0 = kb + 2 * lh;                // lanes 16-31 hold K+2,K+3
        v2f a0, b0;
        a0[0] = Arow[k0];       a0[1] = Arow[k0 + 1];
        b0[0] = Bt[k0 * 16 + ln];
        b0[1] = Bt[(k0 + 1) * 16 + ln];
        acc0 = __builtin_amdgcn_wmma_f32_16x16x4_f32(false, a0, false, b0,
                                                     (short)0, acc0, false, false);
        int k1 = k0 + 4;
        v2f a1, b1;
        a1[0] = Arow[k1];       a1[1] = Arow[k1 + 1];
        b1[0] = Bt[k1 * 16 + ln];
        b1[1] = Bt[(k1 + 1) * 16 + ln];
        acc1 = __builtin_amdgcn_wmma_f32_16x16x4_f32(false, a1, false, b1,
                                                     (short)0, acc1, false, false);
    }
    v8f acc = acc0 + acc1;
#pragma unroll
    for (int v = 0; v < 8; ++v) {
        int m = mt * 16 + v + 8 * lh;        // D layout: VGPR v -> M = v (+8 hi half)
        if (m < 81)
            offs[(size_t)(b * 81 + m) * SP + colbase + ln] = acc[v] + bp[m];
    }
}

// ---------------- deformable sampling record (faithful to source) ----------
// off_sp = &offs[b*81*SP + sp];   h,w,d = sampling location, n = tap 0..26
__device__ __forceinline__ void sample_rec(const float* __restrict__ off_sp,
                                           int h, int w, int d, int n,
                                           float g[4], int idx[4]) {
    float ox = off_sp[(size_t)n * SP];
    float oy = off_sp[(size_t)(N27 + n) * SP];
    float oz = off_sp[(size_t)(2 * N27 + n) * SP];
    float px = (float)(h + (n / 9)) + ox;          // (h+1) + (n/9 - 1)
    float py = (float)(w + ((n / 3) % 3)) + oy;
    float pz = (float)(d + (n % 3)) + oz;
    float fx = floorf(px), fy = floorf(py), fz = floorf(pz);
    // q clamps: x with hp-1=29; y AND z with wp-1=29 (faithful); p clips all 29
    float q0x = fminf(fmaxf(fx, 0.f), 29.f), q1x = fminf(fmaxf(fx + 1.f, 0.f), 29.f);
    float q0y = fminf(fmaxf(fy, 0.f), 29.f), q1y = fminf(fmaxf(fy + 1.f, 0.f), 29.f);
    float q0z = fminf(fmaxf(fz, 0.f), 29.f), q1z = fminf(fmaxf(fz + 1.f, 0.f), 29.f);
    float pcx = fminf(fmaxf(px, 0.f), 29.f);
    float pcy = fminf(fmaxf(py, 0.f), 29.f);
    float pcz = fminf(fmaxf(pz, 0.f), 29.f);
    float wx0 = 1.f + q0x - pcx, wx1 = 1.f + q1x - pcx;
    float wy0 = 1.f + q0y - pcy, wy1 = 1.f + q1y - pcy;
    float wz0 = 1.f + q0z - pcz, wz1 = 1.f + q1z - pcz;
    g[0] = wx0 * wy0 * wz0;   // lt
    g[1] = wx1 * wy1 * wz1;   // rb
    g[2] = wx0 * wy1 * wz1;   // lb  (x from lt, y/z from rb)
    g[3] = wx1 * wy0 * wz0;   // rt
    int i0x = (int)q0x, i1x = (int)q1x;
    int i0y = (int)q0y, i1y = (int)q1y;
    int i0z = (int)q0z, i1z = (int)q1z;
    // faithful flat-index bug: qx*wp + qy + qz
    idx[0] = i0x * 30 + i0y + i0z;
    idx[1] = i1x * 30 + i1y + i1z;
    idx[2] = i0x * 30 + i1y + i1z;
    idx[3] = i1x * 30 + i0y + i0z;
}

// ---------------- pooled mean of x_offset (for second routing) -------------
__global__ void __launch_bounds__(256)
pool_xo_kernel(const float* __restrict__ xp, const float* __restrict__ offs,
               float* __restrict__ pooled) {
    __shared__ float red[256];
    int c = blockIdx.x, b = blockIdx.y, tid = threadIdx.x;
    const float* xc  = xp + (size_t)(b * CIN + c) * XPVOL;
    const float* ob0 = offs + (size_t)(b * 81) * SP;
    float s = 0.f;
    const int TOT = SP * N27;
    for (int rec = tid; rec < TOT; rec += 256) {
        int sp = rec / N27, n = rec - sp * N27;
        int h = sp / 784, w = (sp / 28) % 28, d = sp % 28;
        float g[4]; int ix[4];
        sample_rec(ob0 + sp, h, w, d, n, g, ix);
        s += g[0] * xc[ix[0]] + g[1] * xc[ix[1]] + g[2] * xc[ix[2]] + g[3] * xc[ix[3]];
    }
    red[tid] = s;
    __syncthreads();
    for (int o = 128; o > 0; o >>= 1) {
        if (tid < o) red[tid] += red[tid + o];
        __syncthreads();
    }
    if (tid == 0) pooled[b * CIN + c] = red[0] * (1.f / (float)TOT);
}

// ---------------- final conv: fused deformable-gather GEMM via f32 WMMA ----
// out[64 x 21952] = kern_c[64 x 864] * Bmat, where
// Bmat[k=(c,a,bt,c2)][col=(oh,ow,od)] = xo[c, oh, T/9, od, (T%9)*3+c2], T=84a+3ow+bt
// 4 waves per block (one per 16-row M tile); sampling records AND gathered B
// tile staged once in LDS (each deformable sample computed exactly once).
__global__ void __launch_bounds__(128)
final_gemm_kernel(const float* __restrict__ kern_c, const float* __restrict__ xp,
                  const float* __restrict__ offs, float* __restrict__ out) {
    __shared__ float g_s[16 * 27 * 4];       // 6.9 KB weights
    __shared__ int   i_s[16 * 27 * 4];       // 6.9 KB gather indices
    __shared__ float Bt[KDIM * 16];          // 55.3 KB gathered B tile
    int tid  = threadIdx.x;
    int lane = tid & 31, mt = tid >> 5;      // wave -> M tile 0..3
    int ln = lane & 15, lh = lane >> 4;
    int colbase = blockIdx.x * 16;
    int b = blockIdx.z;
    const float* Xb = xp + (size_t)b * CIN * XPVOL;

    // Phase 1: 16 cols x 27 taps sampling records.
    for (int t = tid; t < 16 * 27; t += 128) {
        int cl = t / 27, rem = t - cl * 27;
        int ka = rem / 9, kbt = (rem / 3) % 3, kc = rem % 3;
        int col = colbase + cl;
        int oh = col / 784, ow = (col / 28) % 28, od = col % 28;
        int T = 84 * ka + 3 * ow + kbt;      // inverse of _reshape_x_offset
        int wi = T / 9, sidx = T - 9 * wi;
        int n  = sidx * 3 + kc;
        int sp = (oh * 28 + wi) * 28 + od;
        float g[4]; int ix[4];
        sample_rec(offs + (size_t)(b * 81) * SP + sp, oh, wi, od, n, g, ix);
#pragma unroll
        for (int q = 0; q < 4; ++q) { g_s[t * 4 + q] = g[q]; i_s[t * 4 + q] = ix[q]; }
    }
    __syncthreads();

    // Phase 2: gather B tile once (4 xp loads + dot per element).
    for (int e = tid; e < KDIM * 16; e += 128) {
        int k = e >> 4, cl = e & 15;
        int c = k / 27, rem = k - c * 27;
        const float* xc = Xb + c * XPVOL;
        int rb = (cl * 27 + rem) * 4;
        Bt[e] = g_s[rb + 0] * xc[i_s[rb + 0]] + g_s[rb + 1] * xc[i_s[rb + 1]]
              + g_s[rb + 2] * xc[i_s[rb + 2]] + g_s[rb + 3] * xc[i_s[rb + 3]];
    }
    __syncthreads();

    // Phase 3: WMMA GEMM from LDS B tile.
    int mrow = mt * 16 + ln;                 // M=64 exactly, no guard needed
    const float* Arow = kern_c + (size_t)(b * OUTC + mrow) * KDIM;

    v8f acc0 = {0.f, 0.f, 0.f, 0.f, 0.f, 0.f, 0.f, 0.f};
    v8f acc1 = {0.f, 0.f, 0.f, 0.f, 0.f, 0.f, 0.f, 0.f};
#pragma unroll 4
    for (int kb = 0; kb < KDIM; kb += 8) {
        int k0 = kb + 2 * lh;
        v2f a0, b0;
        a0[0] = Arow[k0];       a0[1] = Arow[k0 + 1];
        b0[0] = Bt[k0 * 16 + ln];
        b0[1] = Bt[(k0 + 1) * 16 + ln];
        acc0 = __builtin_amdgcn_wmma_f32_16x16x4_f32(false, a0, false, b0,
                                                     (short)0, acc0, false, false);
        int k1 = k0 + 4;
        v2f a1, b1;
        a1[0] = Arow[k1];       a1[1] = Arow[k1 + 1];
        b1[0] = Bt[k1 * 16 + ln];
        b1[1] = Bt[(k1 + 1) * 16 + ln];
        acc1 = __builtin_amdgcn_wmma_f32_16x16x4_f32(false, a1, false, b1,
                                                     (short)0, acc1, false, false);
    }
    v8f acc = acc0 + acc1;
#pragma unroll
    for (int v = 0; v < 8; ++v) {
        int m = mt * 16 + v + 8 * lh;
        out[(size_t)(b * OUTC + m) * SP + colbase + ln] = acc[v];   // bias=None
    }
}

// ---------------------------------------------------------------------------
extern "C" void kernel_launch(void* const* d_in, const int* in_sizes, int n_in,
                              void* d_out, int out_size, void* d_ws, size_t ws_size,
                              hipStream_t stream) {
    (void)in_sizes; (void)n_in; (void)out_size; (void)ws_size;
    const float* x     = (const float*)d_in[0];
    const float* Wp    = (const float*)d_in[1];
    const float* fcp_w = (const float*)d_in[2];
    const float* fcp_b = (const float*)d_in[3];
    const float* bp    = (const float*)d_in[4];
    const float* Wc    = (const float*)d_in[5];
    const float* fcc_w = (const float*)d_in[6];
    const float* fcc_b = (const float*)d_in[7];
    float* out = (float*)d_out;

    // Workspace layout (floats), total ~22.1 MB
    float* ws = (float*)d_ws;
    float* pooled_x = ws;                      // 64
    float* pooled_c = ws + 64;                 // 64
    float* r_p      = ws + 128;                // 16
    float* r_c      = ws + 144;                // 16
    float* kern_p   = ws + 160;                // 2*81*864 = 139968
    float* kern_c   = kern_p + 2 * 81 * KDIM;  // 2*64*864 = 110592
    float* xp       = kern_c + 2 * 64 * KDIM;  // 2*32*27000 = 1728000
    float* offs     = xp + (size_t)BATCH * CIN * XPVOL;   // 2*81*21952 = 3556224

    // 1) routing #1 from x
    pool_x_kernel<<<1, 64, 0, stream>>>(x, pooled_x);
    route_kernel<<<1, 16, 0, stream>>>(pooled_x, fcp_w, fcp_b, r_p);
    // 2) mix p-conv experts
    {
        int total = BATCH * 81 * KDIM;
        mix_kernel<<<(total + 255) / 256, 256, 0, stream>>>(Wp, r_p, kern_p, 81);
    }
    // 3) pad x
    {
        int total = BATCH * CIN * XPVOL;
        pad_kernel<<<(total + 255) / 256, 256, 0, stream>>>(x, xp);
    }
    // 4) offset conv (WMMA implicit-im2col GEMM, LDS-staged B)
    offset_gemm_kernel<<<dim3(SP / 16, 1, BATCH), 192, 0, stream>>>(kern_p, xp, bp, offs);
    // 5) routing #2 from mean(x_offset)
    pool_xo_kernel<<<dim3(CIN, BATCH), 256, 0, stream>>>(xp, offs, pooled_c);
    route_kernel<<<1, 16, 0, stream>>>(pooled_c, fcc_w, fcc_b, r_c);
    // 6) mix final-conv experts
    {
        int total = BATCH * OUTC * KDIM;
        mix_kernel<<<(total + 255) / 256, 256, 0, stream>>>(Wc, r_c, kern_c, OUTC);
    }
    // 7) final fused deformable-gather GEMM (WMMA, LDS-staged B)
    final_gemm_kernel<<<dim3(SP / 16, 1, BATCH), 128, 0, stream>>>(kern_c, xp, offs, out);
}